// GAT_84670985273388
// MI455X (gfx1250) — compile-verified
//
#include <hip/hip_runtime.h>
#include <math.h>

typedef __attribute__((ext_vector_type(2))) float v2f;
typedef __attribute__((ext_vector_type(8))) float v8f;

#define NEG_SLOPE 0.2f
#define BN_EPS 1e-5f

__device__ __forceinline__ float leaky(float v) { return v > 0.f ? v : NEG_SLOPE * v; }

__device__ __forceinline__ void edge_sd(const int* __restrict__ ei, int E, int e, int& s, int& d) {
    if (e < E) { s = ei[e]; d = ei[E + e]; } else { s = e - E; d = e - E; }
}

// ---------------- GEMM1: h1[N,128] = x[N,128] @ W1[128,128]  (fp32 WMMA) ----------------
__global__ __launch_bounds__(256) void k_gemm1(const float* __restrict__ X,
                                               const float* __restrict__ W,
                                               float* __restrict__ Y) {
    __shared__ float lds[16 * 130];            // 130-float row pitch: conflict-free A reads
    const int rowBase = blockIdx.x << 4;
    const int t = threadIdx.x;
#pragma unroll
    for (int j = 0; j < 8; ++j) {              // stage 16x128 x-tile, fully coalesced
        int idx = (j << 8) + t;
        int r = idx >> 7, c = idx & 127;
        lds[r * 130 + c] = X[(size_t)(rowBase + r) * 128 + c];
    }
    __syncthreads();
    const int lane = t & 31;
    const int n = lane & 15;                   // A: row M ; B: col N ; C: col N
    const int g = lane >> 4;                   // lane group selects K pair (0,1)/(2,3)
    const int colBase = (t >> 5) << 4;         // 8 waves -> 8 column tiles
    v8f acc = {0.f, 0.f, 0.f, 0.f, 0.f, 0.f, 0.f, 0.f};
#pragma unroll
    for (int k = 0; k < 128; k += 4) {
        const int ka = k + (g << 1);
        v2f a, b;
        a.x = lds[n * 130 + ka];
        a.y = lds[n * 130 + ka + 1];
        b.x = W[(size_t)ka * 128 + colBase + n];
        b.y = W[(size_t)(ka + 1) * 128 + colBase + n];
        acc = __builtin_amdgcn_wmma_f32_16x16x4_f32(false, a, false, b, (short)0, acc,
                                                    false, false);
    }
#pragma unroll
    for (int r = 0; r < 8; ++r)                // VGPR r: lanes0-15 -> M=r, lanes16-31 -> M=8+r
        Y[(size_t)(rowBase + r + (g << 3)) * 128 + colBase + n] = acc[r];
}

// ---------------- GEMM2: out2[N,48] = h2[N,128] @ W2pad[128,48]  (fp32 WMMA) ----------------
__global__ __launch_bounds__(128) void k_gemm2(const float* __restrict__ X,
                                               const float* __restrict__ Wp,
                                               float* __restrict__ Y) {
    __shared__ float lds[16 * 130];
    const int rowBase = blockIdx.x << 4;
    const int t = threadIdx.x;
#pragma unroll
    for (int j = 0; j < 16; ++j) {
        int idx = (j << 7) + t;
        int r = idx >> 7, c = idx & 127;
        lds[r * 130 + c] = X[(size_t)(rowBase + r) * 128 + c];
    }
    __syncthreads();
    const int wave = t >> 5;
    if (wave < 3) {                            // wave-uniform: EXEC all-ones inside WMMA
        const int lane = t & 31;
        const int n = lane & 15;
        const int g = lane >> 4;
        const int colBase = wave << 4;
        v8f acc = {0.f, 0.f, 0.f, 0.f, 0.f, 0.f, 0.f, 0.f};
#pragma unroll
        for (int k = 0; k < 128; k += 4) {
            const int ka = k + (g << 1);
            v2f a, b;
            a.x = lds[n * 130 + ka];
            a.y = lds[n * 130 + ka + 1];
            b.x = Wp[(size_t)ka * 48 + colBase + n];
            b.y = Wp[(size_t)(ka + 1) * 48 + colBase + n];
            acc = __builtin_amdgcn_wmma_f32_16x16x4_f32(false, a, false, b, (short)0, acc,
                                                        false, false);
        }
#pragma unroll
        for (int r = 0; r < 8; ++r)
            Y[(size_t)(rowBase + r + (g << 3)) * 48 + colBase + n] = acc[r];
    }
}

// ---------------- small helpers ----------------
__global__ void k_padw2(const float* __restrict__ W2, float* __restrict__ Wp) {
    int i = blockIdx.x * blockDim.x + threadIdx.x;
    if (i >= 128 * 48) return;
    int k = i / 48, c = i - k * 48;
    Wp[i] = (c < 40) ? W2[k * 40 + c] : 0.f;
}

// zero agg1, compute per-(node,head) attention dots, init m1/-inf, denom1/0
__global__ void k_prep1(const float* __restrict__ h1, const float* __restrict__ aws,
                        const float* __restrict__ awd, float* __restrict__ agg1,
                        float* __restrict__ as1, float* __restrict__ ad1,
                        float* __restrict__ m1, float* __restrict__ d1, int N) {
    int i = blockIdx.x * blockDim.x + threadIdx.x;
    if (i >= N * 128) return;
    agg1[i] = 0.f;
    if (i < N * 8) {
        int n = i >> 3, h = i & 7;
        const float* row = h1 + (size_t)n * 128 + h * 16;
        float s = 0.f, d = 0.f;
#pragma unroll
        for (int c = 0; c < 16; ++c) {
            float v = row[c];
            s += v * aws[h * 16 + c];
            d += v * awd[h * 16 + c];
        }
        as1[i] = s; ad1[i] = d;
        m1[i] = -INFINITY; d1[i] = 0.f;
    }
}

__global__ void k_edge_max1(const int* __restrict__ ei, int E, int Etot,
                            const float* __restrict__ as1, const float* __restrict__ ad1,
                            float* __restrict__ m1) {
    int i = blockIdx.x * blockDim.x + threadIdx.x;
    if (i >= Etot) return;
    int s, d; edge_sd(ei, E, i, s, d);
#pragma unroll
    for (int h = 0; h < 8; ++h) {
        float e = leaky(as1[s * 8 + h] + ad1[d * 8 + h]);
        atomicMax(&m1[d * 8 + h], e);          // global_atomic_max_num_f32
    }
}

__global__ void k_edge_sum1(const int* __restrict__ ei, int E, int Etot,
                            const float* __restrict__ as1, const float* __restrict__ ad1,
                            const float* __restrict__ m1, float* __restrict__ ee1,
                            float* __restrict__ d1) {
    int i = blockIdx.x * blockDim.x + threadIdx.x;
    if (i >= Etot) return;
    int s, d; edge_sd(ei, E, i, s, d);
#pragma unroll
    for (int h = 0; h < 8; ++h) {
        float e = leaky(as1[s * 8 + h] + ad1[d * 8 + h]);
        float x = __expf(e - m1[d * 8 + h]);
        ee1[(size_t)i * 8 + h] = x;
        atomicAdd(&d1[d * 8 + h], x);
    }
}

// one thread per (edge, channel): 128 channels, head = c>>4. L2-resident random traffic.
__global__ void k_edge_msg1(const int* __restrict__ ei, int E, int Etot,
                            const float* __restrict__ h1, const float* __restrict__ ee1,
                            const float* __restrict__ d1, float* __restrict__ agg1) {
    int i = blockIdx.x * blockDim.x + threadIdx.x;
    if (i >= (Etot << 7)) return;
    int e = i >> 7, c = i & 127, h = c >> 4;
    int s, d; edge_sd(ei, E, e, s, d);
    float alpha = ee1[(size_t)e * 8 + h] / (d1[d * 8 + h] + 1e-16f);
    atomicAdd(&agg1[(size_t)d * 128 + c], h1[(size_t)s * 128 + c] * alpha);
}

__global__ void k_bnelu(float* __restrict__ hh, const float* __restrict__ b1,
                        const float* __restrict__ g, const float* __restrict__ be,
                        const float* __restrict__ mu, const float* __restrict__ var,
                        int total) {
    int i = blockIdx.x * blockDim.x + threadIdx.x;
    if (i >= total) return;
    int c = i & 127;
    float v = hh[i] + b1[c];
    v = (v - mu[c]) * rsqrtf(var[c] + BN_EPS) * g[c] + be[c];
    hh[i] = v > 0.f ? v : (__expf(v) - 1.f);   // ELU(alpha=1)
}

__global__ void k_prep2(const float* __restrict__ out2, const float* __restrict__ aws,
                        const float* __restrict__ awd, float* __restrict__ agg2,
                        float* __restrict__ as2, float* __restrict__ ad2,
                        float* __restrict__ m2, float* __restrict__ d2, int N) {
    int i = blockIdx.x * blockDim.x + threadIdx.x;
    if (i >= N * 40) return;
    agg2[i] = 0.f;
    if (i < N) {
        const float* row = out2 + (size_t)i * 48;
        float s = 0.f, d = 0.f;
#pragma unroll
        for (int c = 0; c < 40; ++c) { s += row[c] * aws[c]; d += row[c] * awd[c]; }
        as2[i] = s; ad2[i] = d; m2[i] = -INFINITY; d2[i] = 0.f;
    }
}

__global__ void k_edge_max2(const int* __restrict__ ei, int E, int Etot,
                            const float* __restrict__ as2, const float* __restrict__ ad2,
                            float* __restrict__ m2) {
    int i = blockIdx.x * blockDim.x + threadIdx.x;
    if (i >= Etot) return;
    int s, d; edge_sd(ei, E, i, s, d);
    atomicMax(&m2[d], leaky(as2[s] + ad2[d]));
}

__global__ void k_edge_sum2(const int* __restrict__ ei, int E, int Etot,
                            const float* __restrict__ as2, const float* __restrict__ ad2,
                            const float* __restrict__ m2, float* __restrict__ ee2,
                            float* __restrict__ d2) {
    int i = blockIdx.x * blockDim.x + threadIdx.x;
    if (i >= Etot) return;
    int s, d; edge_sd(ei, E, i, s, d);
    float x = __expf(leaky(as2[s] + ad2[d]) - m2[d]);
    ee2[i] = x;
    atomicAdd(&d2[d], x);
}

__global__ void k_edge_msg2(const int* __restrict__ ei, int E, int Etot,
                            const float* __restrict__ out2, const float* __restrict__ ee2,
                            const float* __restrict__ d2, float* __restrict__ agg2) {
    int i = blockIdx.x * blockDim.x + threadIdx.x;
    if (i >= Etot * 40) return;
    int e = i / 40, c = i - e * 40;
    int s, d; edge_sd(ei, E, e, s, d);
    float alpha = ee2[e] / (d2[d] + 1e-16f);
    atomicAdd(&agg2[(size_t)d * 40 + c], out2[(size_t)s * 48 + c] * alpha);
}

__global__ void k_lsm(const float* __restrict__ agg2, const float* __restrict__ b2,
                      float* __restrict__ out, int N) {
    int i = blockIdx.x * blockDim.x + threadIdx.x;
    if (i >= N) return;
    float v[40];
    float mx = -INFINITY;
#pragma unroll
    for (int c = 0; c < 40; ++c) { v[c] = agg2[(size_t)i * 40 + c] + b2[c]; mx = fmaxf(mx, v[c]); }
    float s = 0.f;
#pragma unroll
    for (int c = 0; c < 40; ++c) s += __expf(v[c] - mx);
    float l = __logf(s);
#pragma unroll
    for (int c = 0; c < 40; ++c) out[(size_t)i * 40 + c] = v[c] - mx - l;
}

extern "C" void kernel_launch(void* const* d_in, const int* in_sizes, int n_in,
                              void* d_out, int out_size, void* d_ws, size_t ws_size,
                              hipStream_t stream) {
    (void)n_in; (void)out_size; (void)ws_size;
    const float* x    = (const float*)d_in[0];
    const int*   ei   = (const int*)d_in[1];
    const float* W1   = (const float*)d_in[2];
    const float* aws1 = (const float*)d_in[3];
    const float* awd1 = (const float*)d_in[4];
    const float* b1   = (const float*)d_in[5];
    const float* bng  = (const float*)d_in[6];
    const float* bnb  = (const float*)d_in[7];
    const float* bnm  = (const float*)d_in[8];
    const float* bnv  = (const float*)d_in[9];
    const float* W2   = (const float*)d_in[10];
    const float* aws2 = (const float*)d_in[11];
    const float* awd2 = (const float*)d_in[12];
    const float* b2   = (const float*)d_in[13];

    const int N = in_sizes[0] / 128;     // 50000
    const int E = in_sizes[1] / 2;       // 800000
    const int Etot = E + N;              // self-loops appended

    // ---- workspace layout (floats), ~85 MB, layer-2 reuses dead layer-1 regions ----
    float* ws   = (float*)d_ws;
    float* h1   = ws;                               // N*128
    float* agg1 = h1 + (size_t)N * 128;             // N*128 (BN+ELU in place -> h2)
    float* ee1  = agg1 + (size_t)N * 128;           // Etot*8
    float* as1  = ee1 + (size_t)Etot * 8;           // N*8
    float* ad1  = as1 + (size_t)N * 8;              // N*8
    float* m1   = ad1 + (size_t)N * 8;              // N*8
    float* d1   = m1 + (size_t)N * 8;               // N*8
    // layer-2 reuse: h1 region (N*128 floats) holds out2 + agg2 + ee2 (N*88 + Etot < N*128)
    float* out2 = h1;                               // N*48 (padded)
    float* agg2 = out2 + (size_t)N * 48;            // N*40
    float* ee2  = agg2 + (size_t)N * 40;            // Etot
    // layer-2 reuse: ee1 region holds small per-node arrays + padded W2
    float* as2  = ee1;                              // N
    float* ad2  = as2 + N;                          // N
    float* m2   = ad2 + N;                          // N
    float* d2   = m2 + N;                           // N
    float* W2p  = d2 + N;                           // 128*48

    const int TB = 256;
    const int eb = (Etot + TB - 1) / TB;

    k_padw2<<<(128 * 48 + TB - 1) / TB, TB, 0, stream>>>(W2, W2p);
    // ---- layer 1 ----
    k_gemm1<<<N / 16, 256, 0, stream>>>(x, W1, h1);
    k_prep1<<<(N * 128 + TB - 1) / TB, TB, 0, stream>>>(h1, aws1, awd1, agg1, as1, ad1, m1, d1, N);
    k_edge_max1<<<eb, TB, 0, stream>>>(ei, E, Etot, as1, ad1, m1);
    k_edge_sum1<<<eb, TB, 0, stream>>>(ei, E, Etot, as1, ad1, m1, ee1, d1);
    k_edge_msg1<<<((Etot << 7) + TB - 1) / TB, TB, 0, stream>>>(ei, E, Etot, h1, ee1, d1, agg1);
    k_bnelu<<<(N * 128 + TB - 1) / TB, TB, 0, stream>>>(agg1, b1, bng, bnb, bnm, bnv, N * 128);
    // ---- layer 2 ----
    k_gemm2<<<N / 16, 128, 0, stream>>>(agg1, W2p, out2);
    k_prep2<<<(N * 40 + TB - 1) / TB, TB, 0, stream>>>(out2, aws2, awd2, agg2, as2, ad2, m2, d2, N);
    k_edge_max2<<<eb, TB, 0, stream>>>(ei, E, Etot, as2, ad2, m2);
    k_edge_sum2<<<eb, TB, 0, stream>>>(ei, E, Etot, as2, ad2, m2, ee2, d2);
    k_edge_msg2<<<(Etot * 40 + TB - 1) / TB, TB, 0, stream>>>(ei, E, Etot, out2, ee2, d2, agg2);
    k_lsm<<<(N + TB - 1) / TB, TB, 0, stream>>>(agg2, b2, (float*)d_out, N);
}